// AugmentationLayerV2_68779606279024
// MI455X (gfx1250) — compile-verified
//
#include <hip/hip_runtime.h>

typedef __attribute__((ext_vector_type(2))) float v2f;
typedef __attribute__((ext_vector_type(8))) float v8f;

#define SDIM 256
#define CCH  5
#define NPIX (SDIM * SDIM)

static __device__ __forceinline__ float grayw() {
    return (float)(0.2989 + 0.5870 + 0.1140);
}

// Per-image geometry params (uniform within a block; lands in SGPRs).
static __device__ __forceinline__ void geo_params(
    int b, const float* __restrict__ off_frac, const int* __restrict__ crop_size,
    const unsigned char* __restrict__ do_crop,
    float& scale, float& offR, float& offC)
{
    const bool dc = do_crop[b] != 0;
    const float sf = dc ? (float)crop_size[b] : (float)SDIM;
    const float range = (float)SDIM - sf + 1.0f;
    offR = dc ? floorf(off_frac[2 * b + 0] * range) : 0.0f;
    offC = dc ? floorf(off_frac[2 * b + 1] * range) : 0.0f;
    scale = sf * (1.0f / (float)SDIM);
}

static __device__ __forceinline__ void taps(float coord, int& i0, int& i1, float& f) {
    const float cf = floorf(coord);
    f = coord - cf;
    const int c0 = (int)cf;
    i0 = min(max(c0, 0), SDIM - 1);
    i1 = min(max(c0 + 1, 0), SDIM - 1);
}

// ---------------------------------------------------------------------------
// Pass 1: per-(image,channel) sum of the bilinearly resampled plane.
// Separable trick: sum = v^T * M * u with u/v = accumulated tap weights.
// Reads each image exactly once (coalesced). Per-lane accumulation feeds the
// matrix pipe: D += ones(16x4) x B, so sum(D) = 16 * sum(all B contributions),
// independent of the WMMA VGPR layout. One workgroup (1024 thr) per image.
// ---------------------------------------------------------------------------
__global__ __launch_bounds__(1024)
void aug_sum_kernel(const float* __restrict__ crops,
                    const float* __restrict__ off_frac,
                    const int* __restrict__ crop_size,
                    const unsigned char* __restrict__ do_crop,
                    float* __restrict__ sums)
{
    __shared__ float u[SDIM];
    __shared__ float v[SDIM];
    __shared__ float lsum[CCH];

    const int b = blockIdx.x;
    const int tid = threadIdx.x;

    float scale, offR, offC;
    geo_params(b, off_frac, crop_size, do_crop, scale, offR, offC);

    if (tid < SDIM) { u[tid] = 0.0f; v[tid] = 0.0f; }
    if (tid < CCH)  { lsum[tid] = 0.0f; }
    __syncthreads();

    if (tid < SDIM) {
        // column tap weights (axis 1 -> off[1])
        int i0, i1; float f;
        taps(((float)tid + 0.5f) * scale - 0.5f + offC, i0, i1, f);
        atomicAdd(&u[i0], 1.0f - f);
        atomicAdd(&u[i1], f);
    } else if (tid < 2 * SDIM) {
        // row tap weights (axis 0 -> off[0])
        const int r = tid - SDIM;
        int i0, i1; float f;
        taps(((float)r + 0.5f) * scale - 0.5f + offR, i0, i1, f);
        atomicAdd(&v[i0], 1.0f - f);
        atomicAdd(&v[i1], f);
    }
    __syncthreads();

    const float* __restrict__ base = crops + (size_t)b * (size_t)NPIX * CCH;

    const v2f ones = {1.0f, 1.0f};
    v8f acc[CCH];
#pragma unroll
    for (int ch = 0; ch < CCH; ++ch) acc[ch] = (v8f){0, 0, 0, 0, 0, 0, 0, 0};

    // Uniform trip count (65536/1024 == 64) -> EXEC all-ones for WMMA.
    for (int p = tid; p < NPIX; p += 1024) {
        const int rr = p >> 8;
        const int cc = p & (SDIM - 1);
        const float w = v[rr] * u[cc];
        const float* __restrict__ px = base + (size_t)p * CCH;
#pragma unroll
        for (int ch = 0; ch < CCH; ++ch) {
            v2f bop;
            bop.x = w * px[ch];
            bop.y = 0.0f;
            // D = ones(16x4) x B + C ; layout-agnostic reduction on matrix pipe
            acc[ch] = __builtin_amdgcn_wmma_f32_16x16x4_f32(
                false, ones, false, bop, (short)0, acc[ch], false, false);
        }
    }

#pragma unroll
    for (int ch = 0; ch < CCH; ++ch) {
        float t = acc[ch][0] + acc[ch][1] + acc[ch][2] + acc[ch][3]
                + acc[ch][4] + acc[ch][5] + acc[ch][6] + acc[ch][7];
        t *= 0.0625f; // sum(D) = 16 * (this wave's contributions); undo the 16x
        atomicAdd(&lsum[ch], t);
    }
    __syncthreads();

    if (tid < CCH) sums[b * CCH + tid] = lsum[tid];
}

// ---------------------------------------------------------------------------
// Pass 2: per output pixel — invert rot90/flip, 4-tap bilinear gather (input
// is L2-resident), fused brightness/contrast/GRAY. Coalesced 20B stores.
// grid = (256 rows, 64 images), block = 256.
// ---------------------------------------------------------------------------
__global__ __launch_bounds__(256)
void aug_main_kernel(const float* __restrict__ crops,
                     const float* __restrict__ off_frac,
                     const float* __restrict__ bright,
                     const float* __restrict__ contrast,
                     const int* __restrict__ crop_size,
                     const unsigned char* __restrict__ do_crop,
                     const unsigned char* __restrict__ flip,
                     const int* __restrict__ rot_k,
                     const float* __restrict__ sums,
                     float* __restrict__ out)
{
    const int b = blockIdx.y;
    const int i = blockIdx.x;
    const int j = threadIdx.x;

    float scale, offR, offC;
    geo_params(b, off_frac, crop_size, do_crop, scale, offR, offC);

    const int k = rot_k[b] & 3;
    const bool flp = flip[b] != 0;

    // Inverse of rot90(k) (ccw): out[i,j] = pre[r,c]
    int r, c;
    switch (k) {
        case 0: r = i;            c = j;            break;
        case 1: r = j;            c = SDIM - 1 - i; break;
        case 2: r = SDIM - 1 - i; c = SDIM - 1 - j; break;
        default: r = SDIM - 1 - j; c = i;           break;
    }
    if (flp) c = SDIM - 1 - c;  // flip happened before rotation

    int iy0, iy1, ix0, ix1; float fy, fx;
    taps(((float)r + 0.5f) * scale - 0.5f + offR, iy0, iy1, fy);
    taps(((float)c + 0.5f) * scale - 0.5f + offC, ix0, ix1, fx);

    const float w00 = (1.0f - fy) * (1.0f - fx);
    const float w01 = (1.0f - fy) * fx;
    const float w10 = fy * (1.0f - fx);
    const float w11 = fy * fx;

    const float* __restrict__ base = crops + (size_t)b * (size_t)NPIX * CCH;
    const float* __restrict__ p00 = base + (size_t)(iy0 * SDIM + ix0) * CCH;
    const float* __restrict__ p01 = base + (size_t)(iy0 * SDIM + ix1) * CCH;
    const float* __restrict__ p10 = base + (size_t)(iy1 * SDIM + ix0) * CCH;
    const float* __restrict__ p11 = base + (size_t)(iy1 * SDIM + ix1) * CCH;

    float* __restrict__ o = out + (((size_t)b * SDIM + i) * SDIM + j) * CCH;

    const float inv_npix = 1.0f / (float)NPIX;
    const float g = grayw();

#pragma unroll
    for (int ch = 0; ch < CCH; ++ch) {
        const float val = w00 * p00[ch] + w01 * p01[ch]
                        + w10 * p10[ch] + w11 * p11[ch];
        const float br = bright[b * CCH + ch];
        const float ct = contrast[b * CCH + ch];
        const float m  = sums[b * CCH + ch] * inv_npix + br;  // mean(img)+b
        const float xv = val + br;
        o[ch] = ((xv - m) * ct + m) * g;
    }
}

extern "C" void kernel_launch(void* const* d_in, const int* in_sizes, int n_in,
                              void* d_out, int out_size, void* d_ws, size_t ws_size,
                              hipStream_t stream)
{
    (void)in_sizes; (void)n_in; (void)out_size; (void)ws_size;
    const float*         crops     = (const float*)d_in[0];
    const float*         off_frac  = (const float*)d_in[1];
    const float*         bright    = (const float*)d_in[2];
    const float*         contrast  = (const float*)d_in[3];
    const int*           crop_size = (const int*)d_in[4];
    const unsigned char* do_crop   = (const unsigned char*)d_in[5]; // bool (1B)
    const unsigned char* flip      = (const unsigned char*)d_in[6]; // bool (1B)
    const int*           rot_k     = (const int*)d_in[7];

    float* sums = (float*)d_ws;          // 64*5 floats
    float* outp = (float*)d_out;

    aug_sum_kernel<<<64, 1024, 0, stream>>>(crops, off_frac, crop_size, do_crop, sums);
    aug_main_kernel<<<dim3(SDIM, 64), SDIM, 0, stream>>>(
        crops, off_frac, bright, contrast, crop_size, do_crop, flip, rot_k,
        sums, outp);
}